// GNN_node_80977313399680
// MI455X (gfx1250) — compile-verified
//
#include <hip/hip_runtime.h>

#define HIDDEN 256
#define BN_EPS 1e-5f

typedef __attribute__((ext_vector_type(2))) float v2f;
typedef __attribute__((ext_vector_type(8))) float v8f;

__device__ __forceinline__ void atomic_add_f32(float* p, float v) {
  __hip_atomic_fetch_add(p, v, __ATOMIC_RELAXED, __HIP_MEMORY_SCOPE_AGENT);
}

// ---------------------------------------------------------------------------
// Edge aggregation: agg[dst[e]][:] += h[src[e]][:]
// One block per edge, blockDim == feature width (coalesced row gather,
// coalesced atomic scatter). src/dst loads are block-uniform -> scalar loads.
// ---------------------------------------------------------------------------
__global__ void edge_agg_kernel(const float* __restrict__ h,
                                const int* __restrict__ src,
                                const int* __restrict__ dst,
                                float* __restrict__ agg,
                                int nedges, int d) {
  const int e = blockIdx.x;
  if (e >= nedges) return;
  const int s = src[e];
  const int t = dst[e];
  const float v = h[(long)s * d + threadIdx.x];
  atomic_add_f32(&agg[(long)t * d + threadIdx.x], v);
}

// ---------------------------------------------------------------------------
// WMMA fp32 GEMM:  Out[N x 256] = A_op[N x K] * W[K x 256] + bias
//   AOP == 0 : A = A0 + A1                      (h + agg, GIN pre-MLP sum)
//   AOP == 1 : A = relu(fma(A0, scale, shift))  (inner BN + ReLU, fused)
// Block: 256 threads = 8 waves. Block tile 128 rows x 64 cols.
// Each wave: 16 rows x 64 cols = 4 accumulators of v8f.
// W panel staged in LDS in K-pair-interleaved layout:
//   lw[(k>>1)*128 + col*2 + (k&1)]
// so each B fragment (B[ka][col], B[ka+1][col]) is a single aligned
// ds_load_b64 directly into an even-aligned VGPR pair for WMMA.
// Fragment layouts per CDNA5 ISA 7.12.2 (32-bit A 16x4 / C 16x16).
// ---------------------------------------------------------------------------
template <int AOP>
__global__ __launch_bounds__(256)
void gemm_wmma_kernel(const float* __restrict__ A0,
                      const float* __restrict__ A1,
                      const float* __restrict__ scl,
                      const float* __restrict__ shf,
                      const float* __restrict__ W,     // K x HIDDEN row-major
                      const float* __restrict__ bias,  // HIDDEN
                      float* __restrict__ Out,         // Nr x HIDDEN
                      int Nr, int K) {
  __shared__ float lw[256 * 64];  // (K/2) pair-rows x 64 cols x 2
  __shared__ float lsc[256];
  __shared__ float lsh[256];

  const int tid = threadIdx.x;
  const int colbase = blockIdx.x * 64;
  const int rowblock = blockIdx.y * 128;

  // Stage the weight panel, K-pair interleaved (coalesced global reads).
  for (int i = tid; i < K * 64; i += 256) {
    const int kk = i >> 6;
    const int cc = i & 63;
    lw[((kk >> 1) * 64 + cc) * 2 + (kk & 1)] =
        W[(long)kk * HIDDEN + colbase + cc];
  }
  if (AOP == 1) {
    for (int i = tid; i < K; i += 256) {
      lsc[i] = scl[i];
      lsh[i] = shf[i];
    }
  }
  __syncthreads();

  const int wave = tid >> 5;
  const int lane = tid & 31;
  const int half = lane >> 4;   // K-pair selector for A/B frags
  const int l15  = lane & 15;   // M (A) / N (B,C) index

  const int mrow = rowblock + wave * 16 + l15;
  const bool mvalid = mrow < Nr;
  const long arow = (long)(mvalid ? mrow : 0) * K;

  v8f acc0 = {0.f, 0.f, 0.f, 0.f, 0.f, 0.f, 0.f, 0.f};
  v8f acc1 = acc0, acc2 = acc0, acc3 = acc0;

  for (int k0 = 0; k0 < K; k0 += 4) {
    const int ka = k0 + 2 * half;  // this lane supplies K = ka, ka+1

    // A fragment (16x4): lane l15 = row, half selects K pair. float2 load.
    v2f a;
    {
      const float2 p = *(const float2*)(A0 + arow + ka);
      if (AOP == 0) {
        const float2 q = *(const float2*)(A1 + arow + ka);
        a.x = p.x + q.x;
        a.y = p.y + q.y;
      } else {
        a.x = fmaxf(fmaf(p.x, lsc[ka], lsh[ka]), 0.f);
        a.y = fmaxf(fmaf(p.y, lsc[ka + 1], lsh[ka + 1]), 0.f);
      }
      if (!mvalid) { a.x = 0.f; a.y = 0.f; }
    }

    // B fragments: one ds_load_b64 each (K-pair interleaved panel).
    const float* wp = &lw[((ka >> 1) * 64) * 2];
    const v2f b0 = *(const v2f*)(wp + 2 * l15);
    const v2f b1 = *(const v2f*)(wp + 2 * (16 + l15));
    const v2f b2 = *(const v2f*)(wp + 2 * (32 + l15));
    const v2f b3 = *(const v2f*)(wp + 2 * (48 + l15));

    acc0 = __builtin_amdgcn_wmma_f32_16x16x4_f32(false, a, false, b0, (short)0, acc0, false, false);
    acc1 = __builtin_amdgcn_wmma_f32_16x16x4_f32(false, a, false, b1, (short)0, acc1, false, false);
    acc2 = __builtin_amdgcn_wmma_f32_16x16x4_f32(false, a, false, b2, (short)0, acc2, false, false);
    acc3 = __builtin_amdgcn_wmma_f32_16x16x4_f32(false, a, false, b3, (short)0, acc3, false, false);
  }

  // Bias + store. C/D layout: lane -> col l15, VGPR r -> row r + 8*half.
  auto store_tile = [&](v8f av, int t) {
    const int col = colbase + 16 * t + l15;
    const float bv = bias[col];
#pragma unroll
    for (int r = 0; r < 8; ++r) {
      const int row = rowblock + wave * 16 + 8 * half + r;
      if (row < Nr) Out[(long)row * HIDDEN + col] = av[r] + bv;
    }
  };
  store_tile(acc0, 0);
  store_tile(acc1, 1);
  store_tile(acc2, 2);
  store_tile(acc3, 3);
}

// ---------------------------------------------------------------------------
// Per-column sum / sum-of-squares over an N x 256 matrix (for BatchNorm).
// blockDim == 256; thread c owns column c; coalesced row-major reads.
// ---------------------------------------------------------------------------
__global__ void colstats_kernel(const float* __restrict__ t, int n,
                                float* __restrict__ sums,
                                float* __restrict__ sumsq) {
  const int c = threadIdx.x;
  float s = 0.f, s2 = 0.f;
  for (int r = blockIdx.x; r < n; r += gridDim.x) {
    const float v = t[(long)r * HIDDEN + c];
    s += v;
    s2 += v * v;
  }
  atomic_add_f32(&sums[c], s);
  atomic_add_f32(&sumsq[c], s2);
}

// scale = gamma * rsqrt(var + eps), shift = beta - mean*scale
// (biased variance = E[x^2] - E[x]^2, matching jnp.var)
__global__ void bnstats_finalize_kernel(const float* __restrict__ sums,
                                        const float* __restrict__ sumsq,
                                        const float* __restrict__ g,
                                        const float* __restrict__ be, int n,
                                        float* __restrict__ scale,
                                        float* __restrict__ shift) {
  const int c = threadIdx.x;
  const float inv_n = 1.f / (float)n;
  const float mu = sums[c] * inv_n;
  const float var = sumsq[c] * inv_n - mu * mu;
  const float sc = g[c] * rsqrtf(var + BN_EPS);
  scale[c] = sc;
  shift[c] = be[c] - mu * sc;
}

__global__ void bn_apply_kernel(const float* __restrict__ in,
                                float* __restrict__ out,
                                const float* __restrict__ scale,
                                const float* __restrict__ shift,
                                long total, int relu) {
  long i = (long)blockIdx.x * blockDim.x + threadIdx.x;
  const long stride = (long)gridDim.x * blockDim.x;
  for (; i < total; i += stride) {
    const int c = (int)(i & (HIDDEN - 1));
    float v = fmaf(in[i], scale[c], shift[c]);
    out[i] = relu ? fmaxf(v, 0.f) : v;
  }
}

// ---------------------------------------------------------------------------
// Host orchestration.
// Inputs (setup_inputs dict order): x, edge_index,
//   then per layer: w1, b1, g1, be1, w2, b2, g2, be2.
// Workspace: agg/t2 (N*256), t1 (N*256), hbuf (N*256), stats (4*256).
// ---------------------------------------------------------------------------
extern "C" void kernel_launch(void* const* d_in, const int* in_sizes, int n_in,
                              void* d_out, int out_size, void* d_ws,
                              size_t ws_size, hipStream_t stream) {
  (void)in_sizes; (void)n_in; (void)out_size; (void)ws_size;
  const int N = 100000;
  const int E = 500000;
  const int IN_DIM = 128;
  const int NUM_LAYER = 3;

  const float* x = (const float*)d_in[0];
  const int* eidx = (const int*)d_in[1];
  const int* src = eidx;       // edge_index[0]
  const int* dst = eidx + E;   // edge_index[1]

  float* ws = (float*)d_ws;
  const size_t per = (size_t)N * HIDDEN;
  float* agg_t2 = ws;                // aggregation buffer, reused as t2
  float* t1     = ws + per;          // GEMM1 output (pre-BN)
  float* hbuf   = ws + 2 * per;      // layer output h
  float* sums   = ws + 3 * per;
  float* sumsq  = sums + HIDDEN;
  float* scale  = sumsq + HIDDEN;
  float* shift  = scale + HIDDEN;

  const float* hcur = x;
  for (int l = 0; l < NUM_LAYER; ++l) {
    const int K1 = (l == 0) ? IN_DIM : HIDDEN;
    const float* const* p = (const float* const*)(d_in + 2 + 8 * l);
    const float* w1 = p[0];
    const float* b1 = p[1];
    const float* g1 = p[2];
    const float* be1 = p[3];
    const float* w2 = p[4];
    const float* b2 = p[5];
    const float* g2 = p[6];
    const float* be2 = p[7];

    // 1) agg = segment_sum(h[src] -> dst)
    hipMemsetAsync(agg_t2, 0, (size_t)N * K1 * sizeof(float), stream);
    edge_agg_kernel<<<E, K1, 0, stream>>>(hcur, src, dst, agg_t2, E, K1);

    // 2) t1 = (h + agg) @ w1 + b1
    dim3 grid(HIDDEN / 64, (N + 127) / 128);
    gemm_wmma_kernel<0><<<grid, 256, 0, stream>>>(hcur, agg_t2, nullptr,
                                                  nullptr, w1, b1, t1, N, K1);

    // 3) inner BN stats -> scale/shift
    hipMemsetAsync(sums, 0, 2 * HIDDEN * sizeof(float), stream);
    colstats_kernel<<<1024, HIDDEN, 0, stream>>>(t1, N, sums, sumsq);
    bnstats_finalize_kernel<<<1, HIDDEN, 0, stream>>>(sums, sumsq, g1, be1, N,
                                                      scale, shift);

    // 4) t2 = relu(bn(t1)) @ w2 + b2   (BN+ReLU fused into A-fragment load)
    float* t2 = agg_t2;  // agg no longer needed this layer
    gemm_wmma_kernel<1><<<grid, 256, 0, stream>>>(t1, nullptr, scale, shift,
                                                  w2, b2, t2, N, HIDDEN);

    // 5) outer BN (+ ReLU except last layer)
    hipMemsetAsync(sums, 0, 2 * HIDDEN * sizeof(float), stream);
    colstats_kernel<<<1024, HIDDEN, 0, stream>>>(t2, N, sums, sumsq);
    bnstats_finalize_kernel<<<1, HIDDEN, 0, stream>>>(sums, sumsq, g2, be2, N,
                                                      scale, shift);
    float* outbuf = (l == NUM_LAYER - 1) ? (float*)d_out : hbuf;
    bn_apply_kernel<<<2048, 256, 0, stream>>>(t2, outbuf, scale, shift,
                                              (long)N * HIDDEN,
                                              (l != NUM_LAYER - 1) ? 1 : 0);
    hcur = hbuf;
  }
}